// Amplified_PatternMixer_35553739276402
// MI455X (gfx1250) — compile-verified
//
#include <hip/hip_runtime.h>
#include <math.h>

// ---- reference constants -------------------------------------------------
#define NUM_BASIC   5
#define NUM_MIXED   4
#define NUM_FRAME   8
#define NUM_NODES   7
#define NUM_SAMPLES 8
#define MDIM        57                 // 1 + 7*8
#define MMSZ        (MDIM * MDIM)      // 3249
#define CH          256
#define HW          196                // 14*14, == 49 * 4 exactly
#define T2          15                 // 2*NUM_FRAME - 1

typedef __attribute__((ext_vector_type(2))) float v2f;
typedef __attribute__((ext_vector_type(8))) float v8f;

// ---------------------------------------------------------------------------
// Kernel 1: per-row pooled L2 norm via V_WMMA_F32_16X16X4_F32 reduction.
// One block (8 wave32) per row of x [1824 rows x 256ch x 196hw].
// A-matrix tile = 16 channels x 4 spatial elems; B = ones -> D[m,*] = row sum.
// ISA 16x4 f32 A layout: lanes 0-15 carry K=0,1 (VGPR0,1), lanes 16-31 K=2,3.
// ---------------------------------------------------------------------------
__global__ __launch_bounds__(256) void pool_norm_wmma(const float* __restrict__ x,
                                                      float* __restrict__ nrm) {
  const int row  = blockIdx.x;
  const float* xr = x + (size_t)row * (CH * HW);
  const int tid   = threadIdx.x;
  const int wave  = tid >> 5;      // 0..7
  const int lane  = tid & 31;
  const int lhalf = lane >> 4;     // 0 -> K{0,1}, 1 -> K{2,3}
  const int lm    = lane & 15;     // M (channel within 16-wide tile)

  __shared__ float acc;
  if (tid == 0) acc = 0.f;
  __syncthreads();

  v2f ones; ones[0] = 1.f; ones[1] = 1.f;   // B = all ones (layout-invariant)

  float sumsq = 0.f;
  #pragma unroll
  for (int g = 0; g < 2; ++g) {            // each wave covers 2 tiles = 32 ch
    const int ch_base = wave * 32 + g * 16;
    const float* p = xr + (size_t)(ch_base + lm) * HW + lhalf * 2;
    v8f c = {};
    for (int k = 0; k < HW; k += 4) {      // 49 WMMA accumulations
      v2f a = *(const v2f*)(p + k);        // global_load_b64, every byte once
      c = __builtin_amdgcn_wmma_f32_16x16x4_f32(false, a, false, ones,
                                                (short)0, c, false, false);
    }
    // D[j] on lanes 0-15 = chanSum[ch_base+j]; lanes 16-31 = chanSum[ch_base+8+j]
    #pragma unroll
    for (int j = 0; j < 8; ++j) sumsq += c[j] * c[j];
  }
  // combine the two half-tile partials (lane 0 holds ch 0-7, lane 16 ch 8-15)
  sumsq += __shfl_xor(sumsq, 16, 32);
  if (lane == 0) atomicAdd(&acc, sumsq);   // ds_add_f32
  __syncthreads();
  if (tid == 0) nrm[row] = sqrtf(acc) * (1.0f / (float)HW);  // ||mean||_2
}

// ---------------------------------------------------------------------------
// Kernel 2: full _pattern_mixer on one workgroup -> mp[4][57][57]
// ---------------------------------------------------------------------------
__device__ __forceinline__ float mm_val(int h, int r, int c,
                                        const float* __restrict__ mixed_mat,
                                        const float (*sM)[NUM_NODES][T2 * NUM_NODES]) {
  float v = mixed_mat[h * MMSZ + r * MDIM + c];
  if (r >= 1 && c >= 1) {
    const int i = (r - 1) / NUM_NODES;     // sample block of this row
    const int n = (r - 1) % NUM_NODES;
    v += sM[h][n][NUM_NODES * (NUM_SAMPLES - 1 - i) + (c - 1)];
    if ((r % NUM_NODES) == 0) {            // masked rows r = 7,14,...,56
      const int ir = r / NUM_NODES - 1;
      if ((c % NUM_NODES) != 0 && ((c - 1) / NUM_NODES) != ir) v = 0.f;
    }
  }
  return v;
}

__global__ __launch_bounds__(256) void pattern_mixer_kernel(
    const float* __restrict__ mat,   const float* __restrict__ sigma,
    const float* __restrict__ lin_w, const float* __restrict__ lin_b,
    const float* __restrict__ mixed_mat, float* __restrict__ mp) {
  __shared__ float ce[NUM_MIXED][NUM_BASIC][T2];            // 300 f
  __shared__ float sM[NUM_MIXED][NUM_NODES][T2 * NUM_NODES]; // 2940 f
  __shared__ float deg[NUM_MIXED][MDIM];                     // 228 f
  const int tid = threadIdx.x;

  // gaussian temporal expansion -> sigmoid
  for (int idx = tid; idx < NUM_MIXED * NUM_BASIC * T2; idx += blockDim.x) {
    const int h = idx / (NUM_BASIC * T2);
    const int rem = idx % (NUM_BASIC * T2);
    const int b = rem / T2, t = rem % T2;
    const float sg = sigma[h * NUM_BASIC + b];
    const float d  = fabsf((float)t - (float)(NUM_FRAME - 1));
    const float te = (0.3989422804014327f / sg) * expf(-(d * d) / (2.f * sg * sg));
    ce[h][b][t] = 1.f / (1.f + expf(-te));
  }
  __syncthreads();

  // mixed[h,n,(t,m)] = relu( sum_b ce*mat*lin_w + lin_b )
  for (int idx = tid; idx < NUM_MIXED * NUM_NODES * T2 * NUM_NODES; idx += blockDim.x) {
    const int h  = idx / (NUM_NODES * T2 * NUM_NODES);
    const int rm = idx % (NUM_NODES * T2 * NUM_NODES);
    const int n  = rm / (T2 * NUM_NODES);
    const int tm = rm % (T2 * NUM_NODES);
    const int t  = tm / NUM_NODES, m = tm % NUM_NODES;
    float s = lin_b[h];
    #pragma unroll
    for (int b = 0; b < NUM_BASIC; ++b)
      s += ce[h][b][t] * mat[(b * NUM_NODES + n) * NUM_NODES + m] * lin_w[h * NUM_BASIC + b];
    sM[h][n][tm] = fmaxf(s, 0.f);
  }
  __syncthreads();

  // symmetric degree normalization D^-1/2 A D^-1/2 over masked mm
  for (int idx = tid; idx < NUM_MIXED * MDIM; idx += blockDim.x) {
    const int h = idx / MDIM, i = idx % MDIM;
    float s = 0.f;
    for (int j = 0; j < MDIM; ++j) s += mm_val(h, i, j, mixed_mat, sM);
    deg[h][i] = 1.f / sqrtf(fmaxf(s, 1.f));
  }
  __syncthreads();

  for (int idx = tid; idx < NUM_MIXED * MMSZ; idx += blockDim.x) {
    const int h  = idx / MMSZ;
    const int rm = idx % MMSZ;
    const int i  = rm / MDIM, j = rm % MDIM;
    mp[idx] = deg[h][i] * mm_val(h, i, j, mixed_mat, sM) * deg[h][j];
  }
}

// ---------------------------------------------------------------------------
// Kernel 3: out[b,h,i,j] = mp[h,i,j] + alpha[h] * nrm[b,j]
// ---------------------------------------------------------------------------
__global__ __launch_bounds__(256) void combine_kernel(
    const float* __restrict__ mp, const float* __restrict__ nrm,
    const float* __restrict__ alpha, float* __restrict__ out, int total) {
  const int idx = blockIdx.x * blockDim.x + threadIdx.x;
  if (idx >= total) return;
  const int per_b = NUM_MIXED * MMSZ;
  const int b   = idx / per_b;
  const int rem = idx % per_b;
  const int h   = rem / MMSZ;
  const int ij  = rem % MMSZ;
  const int j   = ij % MDIM;
  out[idx] = mp[h * MMSZ + ij] + alpha[h] * nrm[b * MDIM + j];
}

// ---------------------------------------------------------------------------
extern "C" void kernel_launch(void* const* d_in, const int* in_sizes, int n_in,
                              void* d_out, int out_size, void* d_ws, size_t ws_size,
                              hipStream_t stream) {
  const float* mat       = (const float*)d_in[0];
  const float* x         = (const float*)d_in[1];
  const float* sigma     = (const float*)d_in[2];
  const float* lin_w     = (const float*)d_in[3];
  const float* lin_b     = (const float*)d_in[4];
  const float* mixed_mat = (const float*)d_in[5];
  const float* alpha     = (const float*)d_in[6];
  float* out = (float*)d_out;

  const int rows = in_sizes[1] / (CH * HW);   // B*M = 1824
  const int B    = rows / MDIM;               // 32

  float* nrm = (float*)d_ws;                  // rows floats
  float* mp  = nrm + 2048;                    // 4*57*57 floats (8KB offset)

  pool_norm_wmma<<<rows, 256, 0, stream>>>(x, nrm);
  pattern_mixer_kernel<<<1, 256, 0, stream>>>(mat, sigma, lin_w, lin_b, mixed_mat, mp);

  const int total = B * NUM_MIXED * MMSZ;     // 415,872
  combine_kernel<<<(total + 255) / 256, 256, 0, stream>>>(mp, nrm, alpha, out, total);
}